// DeformableConvBlock_74981539054069
// MI455X (gfx1250) — compile-verified
//
#include <hip/hip_runtime.h>
#include <hip/hip_bf16.h>

#define KSZ   3
#define PAD   1
#define CIN   64
#define COUT  64
#define BB    4
#define HH    128
#define WW    128
#define KK    9      // KSZ*KSZ
#define NOFF  18     // 2*KK
#define HW    (HH*WW)

typedef __attribute__((ext_vector_type(2))) float v2f;
typedef __attribute__((ext_vector_type(4))) float v4f;
typedef __attribute__((ext_vector_type(8))) float v8f;

// ---------------------------------------------------------------------------
// Kernel 1: offset conv  off[b,ch,h,w] = b_off[ch] + sum_{c,ky,kx} x * w_off
// ---------------------------------------------------------------------------
__global__ void __launch_bounds__(256)
deform_off_conv_kernel(const float* __restrict__ x,
                       const float* __restrict__ w_off,
                       const float* __restrict__ b_off,
                       float* __restrict__ off) {
    int idx = blockIdx.x * blockDim.x + threadIdx.x;           // B*18*H*W threads
    if (idx >= BB * NOFF * HW) return;
    int w  = idx & (WW - 1);
    int h  = (idx >> 7) & (HH - 1);
    int ch = (idx >> 14) % NOFF;
    int b  = idx / (NOFF * HW);

    float acc = b_off[ch];
    const float* xb = x + b * CIN * HW;
    for (int ky = 0; ky < KSZ; ++ky) {
        int yy = h - PAD + ky;
        if (yy < 0 || yy >= HH) continue;
        for (int kx = 0; kx < KSZ; ++kx) {
            int xx = w - PAD + kx;
            if (xx < 0 || xx >= WW) continue;
            const float* xp = xb + yy * WW + xx;
            const float* wp = w_off + ch * CIN * KK + ky * KSZ + kx;
            float a = 0.f;
            #pragma unroll 4
            for (int c = 0; c < CIN; ++c)
                a = fmaf(xp[c * HW], wp[c * KK], a);
            acc += a;
        }
    }
    off[idx] = acc;
}

// ---------------------------------------------------------------------------
// Kernel 1b: pre-pack w_def into WMMA A-fragment order.
// Layout: wpk[((g*72 + s2)*32 + lane)*4 + e]
//   g    = COUT row-group (co_base = g*16)
//   s2   = step pair, step s = 2*s2 + (e>>1), with s = k*16 + c4step
//   lane = m (A row, lane&15) + ksel (lane>>4 selects K-slots {2,3})
//   e    : {a.x(sA), a.y(sA), a.x(sB), a.y(sB)}
// Consumer loads one coalesced b128 per lane per 2 WMMAs.
// ---------------------------------------------------------------------------
__global__ void __launch_bounds__(256)
repack_wdef_kernel(const float* __restrict__ w_def, float* __restrict__ wpk) {
    int j = blockIdx.x * blockDim.x + threadIdx.x;   // 64*576 = 36864 elements
    if (j >= COUT * CIN * KK) return;
    int g    = j / 9216;          // 72*32*4 per row-group
    int rest = j % 9216;
    int s2   = rest >> 7;         // /128
    int l    = (rest >> 2) & 31;
    int e    = rest & 3;

    int s      = s2 * 2 + (e >> 1);
    int k      = s >> 4;
    int c4step = s & 15;
    int m      = l & 15;
    int ksel   = l >> 4;
    int cidx   = c4step * 4 + 2 * ksel + (e & 1);
    int co     = g * 16 + m;
    wpk[j] = w_def[co * (CIN * KK) + cidx * KK + k];
}

// ---------------------------------------------------------------------------
// Kernel 2: fused bilinear sampling + 64x576x16 GEMM via V_WMMA_F32_16X16X4_F32
// One block (128 thr = 4 waves) computes all 64 COUT for 16 consecutive pixels.
// ---------------------------------------------------------------------------
__global__ void __launch_bounds__(128)
deform_wmma_kernel(const float* __restrict__ x,
                   const float* __restrict__ off,
                   const float* __restrict__ wpk,
                   float* __restrict__ out) {
    // sampled values interleaved [k][c/2][p][2] -> ds_load_b64 B fragments
    __shared__ float s_val[KK * CIN * 16];           // 36864 B
    __shared__ int   s_i00[KK * 16], s_i01[KK * 16],
                     s_i10[KK * 16], s_i11[KK * 16]; // corner indices (clamped)
    __shared__ float s_w00[KK * 16], s_w01[KK * 16],
                     s_w10[KK * 16], s_w11[KK * 16]; // validity-folded weights

    const int t   = blockIdx.x;          // 4096 tiles
    const int b   = t >> 10;             // 1024 tiles per batch image
    const int rem = t & 1023;
    const int h   = rem >> 3;            // 8 w-tiles per row
    const int w0  = (rem & 7) << 4;      // 16 pixels per tile
    const int tid = threadIdx.x;

    // ---- stage 1: bilinear parameters per (k, pixel) ----
    for (int kp = tid; kp < KK * 16; kp += 128) {
        int k = kp >> 4;
        int p = kp & 15;
        int ky = k / KSZ, kx = k % KSZ;
        int wpix = w0 + p;
        float dy = off[((b * NOFF + 2 * k    ) * HH + h) * WW + wpix];
        float dx = off[((b * NOFF + 2 * k + 1) * HH + h) * WW + wpix];
        float py = (float)(h    - PAD + ky) + dy;
        float px = (float)(wpix - PAD + kx) + dx;
        float y0f = floorf(py), x0f = floorf(px);
        float fy = py - y0f,    fx = px - x0f;
        int y0 = (int)y0f, x0 = (int)x0f;
        int y1 = y0 + 1,   x1 = x0 + 1;

        bool vy0 = (y0 >= 0) && (y0 < HH);
        bool vy1 = (y1 >= 0) && (y1 < HH);
        bool vx0 = (x0 >= 0) && (x0 < WW);
        bool vx1 = (x1 >= 0) && (x1 < WW);
        int cy0 = min(max(y0, 0), HH - 1), cy1 = min(max(y1, 0), HH - 1);
        int cx0 = min(max(x0, 0), WW - 1), cx1 = min(max(x1, 0), WW - 1);

        s_i00[kp] = cy0 * WW + cx0;
        s_i01[kp] = cy0 * WW + cx1;
        s_i10[kp] = cy1 * WW + cx0;
        s_i11[kp] = cy1 * WW + cx1;
        s_w00[kp] = (vy0 && vx0) ? (1.f - fy) * (1.f - fx) : 0.f;
        s_w01[kp] = (vy0 && vx1) ? (1.f - fy) * fx         : 0.f;
        s_w10[kp] = (vy1 && vx0) ? fy * (1.f - fx)         : 0.f;
        s_w11[kp] = (vy1 && vx1) ? fy * fx                 : 0.f;
    }
    __syncthreads();

    // ---- stage 2: gather + bilinear combine, two channels per thread ----
    for (int i = tid; i < (KK * CIN * 16) / 2; i += 128) {   // 4608 c-pairs
        int p  = i & 15;
        int c2 = (i >> 4) & 31;          // channel-pair index
        int k  = i >> 9;
        int kp = k * 16 + p;
        int i00 = s_i00[kp], i01 = s_i01[kp], i10 = s_i10[kp], i11 = s_i11[kp];
        float w00 = s_w00[kp], w01 = s_w01[kp], w10 = s_w10[kp], w11 = s_w11[kp];
        const float* xc0 = x + (b * CIN + c2 * 2) * HW;
        const float* xc1 = xc0 + HW;
        float v0 = fmaf(w00, xc0[i00], fmaf(w01, xc0[i01],
                   fmaf(w10, xc0[i10], w11 * xc0[i11])));
        float v1 = fmaf(w00, xc1[i00], fmaf(w01, xc1[i01],
                   fmaf(w10, xc1[i10], w11 * xc1[i11])));
        v2f pair = {v0, v1};
        *(v2f*)(s_val + k * (CIN * 16) + c2 * 32 + p * 2) = pair;
    }
    __syncthreads();

    // ---- stage 3: per-wave 16x16 WMMA tile, K = 576 in steps of 4 ----
    const int wid  = tid >> 5;
    const int lane = tid & 31;
    const int m    = lane & 15;   // A: COUT row within tile; B/D: pixel col
    const int ksel = lane >> 4;   // 0: K-slots {0,1}, 1: K-slots {2,3}

    v8f acc = {};
    for (int k = 0; k < KK; ++k) {
        const float* svk = s_val + k * (CIN * 16) + ksel * 32 + 2 * m;
        const float* wq  = wpk + ((wid * 72 + k * 8) * 32 + lane) * 4;
        #pragma unroll
        for (int s2i = 0; s2i < 8; ++s2i) {          // 2 WMMA steps per iter
            v4f w4 = *(const v4f*)(wq + s2i * 128);  // coalesced b128
            v2f a0 = {w4.x, w4.y};
            v2f a1 = {w4.z, w4.w};
            // B frag pairs: c4a = s2i*8, c4b = s2i*8+4 -> c2 offsets 4*s2i(+2)
            v2f b0 = *(const v2f*)(svk + (s2i * 4    ) * 32);  // ds_load_b64
            v2f b1 = *(const v2f*)(svk + (s2i * 4 + 2) * 32);
            acc = __builtin_amdgcn_wmma_f32_16x16x4_f32(
                      false, a0, false, b0, (short)0, acc, false, false);
            acc = __builtin_amdgcn_wmma_f32_16x16x4_f32(
                      false, a1, false, b1, (short)0, acc, false, false);
        }
    }

    // ---- store D: VGPR r holds rows {r, r+8}; lane half selects which ----
    #pragma unroll
    for (int r = 0; r < 8; ++r) {
        int co = (wid << 4) + r + (ksel << 3);
        out[((b * COUT + co) * HH + h) * WW + w0 + m] = acc[r];
    }
}

// ---------------------------------------------------------------------------
extern "C" void kernel_launch(void* const* d_in, const int* in_sizes, int n_in,
                              void* d_out, int out_size, void* d_ws, size_t ws_size,
                              hipStream_t stream) {
    const float* x     = (const float*)d_in[0];   // (4,64,128,128)
    const float* w_off = (const float*)d_in[1];   // (18,64,3,3)
    const float* b_off = (const float*)d_in[2];   // (18,)
    const float* w_def = (const float*)d_in[3];   // (64,64,3,3)
    float* out = (float*)d_out;                   // (4,64,128,128)

    float* off = (float*)d_ws;                    // (4,18,128,128) = 4.5 MB
    float* wpk = off + (size_t)BB * NOFF * HW;    // packed w_def, 147 KB

    const int n_off = BB * NOFF * HW;             // 1,179,648
    deform_off_conv_kernel<<<(n_off + 255) / 256, 256, 0, stream>>>(
        x, w_off, b_off, off);

    const int n_wdef = COUT * CIN * KK;           // 36,864
    repack_wdef_kernel<<<(n_wdef + 255) / 256, 256, 0, stream>>>(w_def, wpk);

    const int n_tiles = BB * HH * (WW / 16);      // 4096 blocks
    deform_wmma_kernel<<<n_tiles, 128, 0, stream>>>(
        x, off, wpk, out);
}